// HierarchicalContrastiveLossNew_41944650613366
// MI455X (gfx1250) — compile-verified
//
#include <hip/hip_runtime.h>
#include <hip/hip_bf16.h>

// ---------------------------------------------------------------------------
// Hierarchical contrastive loss, B=8, T=512, C=320, fp32.
// Temporal loss Gram matrices computed with V_WMMA_F32_16X16X4_F32.
// ---------------------------------------------------------------------------

typedef __attribute__((ext_vector_type(2))) float v2f;
typedef __attribute__((ext_vector_type(8))) float v8f;

#define BB 8
#define CC 320
#define CP 322   // padded LDS row stride (floats); keeps 8B alignment, avoids bank clash

// ---------------------------------------------------------------------------
__global__ void init_acc_kernel(float* __restrict__ o) {
    if (threadIdx.x == 0 && blockIdx.x == 0) o[0] = 0.0f;
}

// ---------------------------------------------------------------------------
// Max-pool z over time pairs: out[b,t,c] = max(in[b,2t,c], in[b,2t+1,c])
__global__ void pool_z_kernel(const float* __restrict__ in, float* __restrict__ outp,
                              int Tin, int T2) {
    const int total = BB * T2 * CC;
    for (int i = blockIdx.x * blockDim.x + threadIdx.x; i < total;
         i += gridDim.x * blockDim.x) {
        const int c  = i % CC;
        const int rs = i / CC;
        const int t2 = rs % T2;
        const int bq = rs / T2;
        const float* base = in + ((size_t)bq * Tin + 2 * t2) * CC + c;
        outp[i] = fmaxf(base[0], base[CC]);
    }
}

// Mean-pool t over time pairs
__global__ void pool_t_kernel(const float* __restrict__ in, float* __restrict__ outp,
                              int Tin, int T2) {
    const int total = BB * T2;
    for (int i = blockIdx.x * blockDim.x + threadIdx.x; i < total;
         i += gridDim.x * blockDim.x) {
        const int t2 = i % T2;
        const int bq = i / T2;
        const float* base = in + (size_t)bq * Tin + 2 * t2;
        outp[i] = 0.5f * (base[0] + base[1]);
    }
}

// ---------------------------------------------------------------------------
// Instance loss: one wave per row i of N = 8*T rows.
//   top-8 smallest |t_i - t_j| (same-seq entries penalized by range+0.001),
//   17 logits (pos, 8x neg1, 8x neg2), -log_softmax[0], atomicAdd scaled.
__global__ __launch_bounds__(256) void instance_kernel(
    const float* __restrict__ z1, const float* __restrict__ z2,
    const float* __restrict__ t, int T, float scale, float* __restrict__ acc)
{
    __shared__ float cd[8][256];
    __shared__ int   ci[8][256];
    __shared__ int   sel[8][8];

    const int lane  = threadIdx.x & 31;
    const int wslot = threadIdx.x >> 5;
    const int i     = blockIdx.x * 8 + wslot;       // row == wave id
    const int N     = BB * T;
    if (i >= N) return;                             // wave-uniform exit, no barriers used
    const int   si  = i / T;
    const float tiv = t[i];

    // pass 1: global range of t -> penalty
    float tmx = -3.0e38f, tmn = 3.0e38f;
    for (int j = lane; j < N; j += 32) { float v = t[j]; tmx = fmaxf(tmx, v); tmn = fminf(tmn, v); }
#pragma unroll
    for (int m2 = 16; m2 >= 1; m2 >>= 1) {
        tmx = fmaxf(tmx, __shfl_xor(tmx, m2, 32));
        tmn = fminf(tmn, __shfl_xor(tmn, m2, 32));
    }
    const float pen = (tmx - tmn) + 0.001f;

    // pass 2: per-lane sorted top-8 (ascending), fully unrolled push-insert
    float bd[8]; int bi[8];
#pragma unroll
    for (int p = 0; p < 8; ++p) { bd[p] = 3.0e38f; bi[p] = 0; }
    for (int j = lane; j < N; j += 32) {
        float dv = fabsf(t[j] - tiv);
        if (j / T == si) dv += pen;
        int jj = j;
#pragma unroll
        for (int p = 0; p < 8; ++p) {
            if (dv < bd[p]) {
                float od = bd[p]; int oi = bi[p];
                bd[p] = dv; bi[p] = jj; dv = od; jj = oi;
            }
        }
    }

    // merge 32 lanes x 8 candidates in LDS; lane 0 selects global top-8
#pragma unroll
    for (int p = 0; p < 8; ++p) { cd[wslot][lane * 8 + p] = bd[p]; ci[wslot][lane * 8 + p] = bi[p]; }
    asm volatile("s_wait_dscnt 0" ::: "memory");
    if (lane == 0) {
        float md[8]; int mi[8];
#pragma unroll
        for (int p = 0; p < 8; ++p) { md[p] = 3.0e38f; mi[p] = 0; }
        for (int q = 0; q < 256; ++q) {
            float dv = cd[wslot][q]; int jj = ci[wslot][q];
#pragma unroll
            for (int p = 0; p < 8; ++p) {
                if (dv < md[p]) {
                    float od = md[p]; int oi = mi[p];
                    md[p] = dv; mi[p] = jj; dv = od; jj = oi;
                }
            }
        }
#pragma unroll
        for (int p = 0; p < 8; ++p) sel[wslot][p] = mi[p];
    }
    asm volatile("s_wait_dscnt 0" ::: "memory");
    int idx[8];
#pragma unroll
    for (int p = 0; p < 8; ++p) idx[p] = sel[wslot][p];

    // 17 logits: each logit is a 320-dot, lane-strided + wave reduce
    float q[10];
#pragma unroll
    for (int u = 0; u < 10; ++u) q[u] = z1[(size_t)i * CC + lane + 32 * u];

    float lg[17];
    {
        float p = 0.0f;
#pragma unroll
        for (int u = 0; u < 10; ++u) p += q[u] * z2[(size_t)i * CC + lane + 32 * u];
#pragma unroll
        for (int m2 = 16; m2 >= 1; m2 >>= 1) p += __shfl_xor(p, m2, 32);
        lg[0] = p;
    }
#pragma unroll
    for (int k = 0; k < 8; ++k) {
        const size_t jb = (size_t)idx[k] * CC;
        float p1 = 0.0f, p2 = 0.0f;
#pragma unroll
        for (int u = 0; u < 10; ++u) {
            const float a1 = z1[jb + lane + 32 * u];
            const float a2 = z2[jb + lane + 32 * u];
            p1 += q[u] * a1; p2 += q[u] * a2;
        }
#pragma unroll
        for (int m2 = 16; m2 >= 1; m2 >>= 1) {
            p1 += __shfl_xor(p1, m2, 32);
            p2 += __shfl_xor(p2, m2, 32);
        }
        lg[1 + k] = p1; lg[9 + k] = p2;
    }

    float mx = lg[0];
#pragma unroll
    for (int k = 1; k < 17; ++k) mx = fmaxf(mx, lg[k]);
    float se = 0.0f;
#pragma unroll
    for (int k = 0; k < 17; ++k) se += expf(lg[k] - mx);
    const float val = (mx + logf(se)) - lg[0];
    if (lane == 0) atomicAdd(acc, scale * val);
}

// ---------------------------------------------------------------------------
// Temporal loss: 2 waves per block; each wave owns one 16-row strip of one
// batch's (2T)x(2T) Gram matrix; the two waves share each staged B tile.
// Per-lane online logsumexp with a branchless fast path for interior tiles;
// masked slow path only for diagonal / pair-column / tail tiles.
// Gram tiles via WMMA f32 16x16x4.
__global__ __launch_bounds__(64) void temporal_kernel(
    const float* __restrict__ z1, const float* __restrict__ z2,
    int T, float scale, float* __restrict__ acc)
{
    __shared__ float As[2][16 * CP];
    __shared__ float Bs[16 * CP];

    const int n      = 2 * T;
    const int strips = (n + 15) >> 4;
    const int spb    = (strips + 1) >> 1;           // strip-pairs per batch
    const int b      = blockIdx.x / spb;
    const int bs     = blockIdx.x % spb;
    const int tid    = threadIdx.x;                 // 0..63
    const int wave   = tid >> 5;
    const int lane   = tid & 31;
    const int half   = lane >> 4;
    const int l16    = lane & 15;

    const int st     = bs * 2 + wave;
    const bool active = (st < strips);
    const int i0     = (active ? st : (strips - 1)) << 4;

    // first pair-column of this strip; 16-aligned whenever T is a multiple of 16
    int pt0 = i0 + T; if (pt0 >= n) pt0 -= n;

    // stage this wave's A strip (16 x 320, row-clamped) in LDS, float2 loads
    float* Aw = &As[wave][0];
    for (int r = 0; r < 16; ++r) {
        int gi = i0 + r; if (gi > n - 1) gi = n - 1;
        const float* src = (gi < T) ? (z1 + ((size_t)b * T + gi) * CC)
                                    : (z2 + ((size_t)b * T + (gi - T)) * CC);
#pragma unroll
        for (int c = lane * 2; c < CC; c += 64)
            *(v2f*)(&Aw[r * CP + c]) = *(const v2f*)(src + c);
    }

    // per-lane online LSE state: lane owns column slot l16 of every tile
    float mr[8], sr[8], pr[8];
#pragma unroll
    for (int r = 0; r < 8; ++r) { mr[r] = -1.0e30f; sr[r] = 0.0f; pr[r] = 0.0f; }

    for (int j0 = 0; j0 < n; j0 += 16) {
        __syncthreads();
        // stage B tile cooperatively (64 threads, float2): 16 rows x 160 float2
        for (int q = tid; q < 16 * 160; q += 64) {
            const int r  = q / 160;
            const int c  = (q - r * 160) * 2;
            int gj = j0 + r; if (gj > n - 1) gj = n - 1;
            const float* src = (gj < T) ? (z1 + ((size_t)b * T + gj) * CC)
                                        : (z2 + ((size_t)b * T + (gj - T)) * CC);
            *(v2f*)(&Bs[r * CP + c]) = *(const v2f*)(src + c);
        }
        __syncthreads();
        asm volatile("s_wait_dscnt 0" ::: "memory");

        // prefetch next B tile (16 rows x 10 cachelines of 128B)
        if (j0 + 16 < n) {
#pragma unroll
            for (int q5 = 0; q5 < 3; ++q5) {
                const int q = q5 * 64 + tid;
                if (q < 160) {
                    const int rr  = q / 10;
                    const int cc0 = (q - rr * 10) * 32;
                    int gj = j0 + 16 + rr; if (gj > n - 1) gj = n - 1;
                    const float* src = (gj < T) ? (z1 + ((size_t)b * T + gj) * CC)
                                                : (z2 + ((size_t)b * T + (gj - T)) * CC);
                    __builtin_prefetch(src + cc0, 0, 3);
                }
            }
        }

        // 16x16 Gram tile: D = A(16xK) * B(Kx16), K-loop over 320 in steps of 4.
        // A frag: lane holds M=l16, K = kk + 2*half (+1). B frag mirrored (N=l16).
        v8f d = {0.f, 0.f, 0.f, 0.f, 0.f, 0.f, 0.f, 0.f};
#pragma unroll 8
        for (int kk = 0; kk < CC; kk += 4) {
            v2f a  = *(const v2f*)(&Aw[l16 * CP + kk + 2 * half]);
            v2f bf = *(const v2f*)(&Bs[l16 * CP + kk + 2 * half]);
            d = __builtin_amdgcn_wmma_f32_16x16x4_f32(
                    false, a, false, bf, (short)0, d, false, false);
        }

        // D layout: element r -> (row = i0 + r + 8*half, col = j0 + l16)
        const bool slowTile = (n < 16) || (j0 == i0) || (j0 == pt0);
        if (!slowTile) {
            // interior tile: all 16 cols valid, no diagonal, no pair column
#pragma unroll
            for (int r = 0; r < 8; ++r) {
                const float v  = d[r];
                const float mn = fmaxf(mr[r], v);
                sr[r] = sr[r] * __expf(mr[r] - mn) + __expf(v - mn);
                mr[r] = mn;
            }
        } else {
#pragma unroll
            for (int r = 0; r < 8; ++r) {
                const int grow = i0 + r + 8 * half;
                const int gcol = j0 + l16;
                const float v  = d[r];
                int pcol = grow + T; if (pcol >= n) pcol -= n;
                const bool valid = (grow < n) && (gcol < n) && (gcol != grow);
                if (valid) {
                    if (gcol == pcol) pr[r] += v;
                    const float mn = fmaxf(mr[r], v);
                    sr[r] = sr[r] * __expf(mr[r] - mn) + __expf(v - mn);
                    mr[r] = mn;
                }
            }
        }
    }

    // merge per-lane (m, s) and pair sums across the 16 lanes of each half
    float tot = 0.0f;
#pragma unroll
    for (int r = 0; r < 8; ++r) {
        float m = mr[r], s = sr[r], pv = pr[r];
#pragma unroll
        for (int m2 = 8; m2 >= 1; m2 >>= 1) {
            const float om = __shfl_xor(m, m2, 16);
            const float os = __shfl_xor(s, m2, 16);
            pv            += __shfl_xor(pv, m2, 16);
            const float mn = fmaxf(m, om);
            s = s * __expf(m - mn) + os * __expf(om - mn);
            m = mn;
        }
        const int grow = i0 + r + 8 * half;
        if (grow < n) tot += (m + logf(s)) - pv;
    }
    if (active && l16 == 0) atomicAdd(acc, scale * tot);  // lanes 0/16: rows 0-7 / 8-15
}

// ---------------------------------------------------------------------------
extern "C" void kernel_launch(void* const* d_in, const int* in_sizes, int n_in,
                              void* d_out, int out_size, void* d_ws, size_t ws_size,
                              hipStream_t stream) {
    const float* cz1 = (const float*)d_in[0];
    const float* cz2 = (const float*)d_in[1];
    const float* ct  = (const float*)d_in[2];
    float* out = (float*)d_out;
    float* w   = (float*)d_ws;

    init_acc_kernel<<<1, 32, 0, stream>>>(out);

    size_t off = 0;
    int T = 512;
    const float DTOT = 10.0f;   // 9 pooled levels + final T==1 level

    for (int d = 0; d < 10; ++d) {
        const int N = BB * T;
        const int iblocks = (N + 7) / 8;   // 8 waves/block, 1 wave per row
        instance_kernel<<<iblocks, 256, 0, stream>>>(
            cz1, cz2, ct, T, 0.5f / ((float)N * DTOT), out);

        if (T == 1) break;

        const int n = 2 * T;
        const int strips = (n + 15) >> 4;
        const int spb = (strips + 1) >> 1;
        temporal_kernel<<<BB * spb, 64, 0, stream>>>(
            cz1, cz2, T, 0.5f / ((float)n * (float)BB * DTOT), out);

        // pool to next level (workspace)
        const int T2 = T / 2;
        float* nz1 = w + off; off += (size_t)BB * T2 * CC;
        float* nz2 = w + off; off += (size_t)BB * T2 * CC;
        float* nt  = w + off; off += (size_t)BB * T2;

        const int totz = BB * T2 * CC;
        pool_z_kernel<<<(totz + 255) / 256, 256, 0, stream>>>(cz1, nz1, T, T2);
        pool_z_kernel<<<(totz + 255) / 256, 256, 0, stream>>>(cz2, nz2, T, T2);
        pool_t_kernel<<<(BB * T2 + 255) / 256, 256, 0, stream>>>(ct, nt, T, T2);

        cz1 = nz1; cz2 = nz2; ct = nt; T = T2;
    }
}